// RelationAwareLinear_52415780880536
// MI455X (gfx1250) — compile-verified
//
#include <hip/hip_runtime.h>

typedef __attribute__((ext_vector_type(2))) float v2f;
typedef __attribute__((ext_vector_type(8))) float v8f;

#define N_ROWS 16384
#define N_REL  64
#define DIM    128
#define MAX_TILES (N_REL + N_ROWS / 16)   // 1088

// ---------------------------------------------------------------------------
// Kernel A: bucket rows by relation, build compact tile worklist in d_ws.
// Single block, 1024 threads. Deterministic final output (each row owns its
// output slot; intra-bucket order is irrelevant).
// ---------------------------------------------------------------------------
__global__ void __launch_bounds__(1024)
rel_bucket_kernel(const int* __restrict__ rel,
                  int* __restrict__ nTiles,
                  int4* __restrict__ tiles,
                  int* __restrict__ rowIdx) {
    __shared__ int lc[N_REL];
    __shared__ int lofs[N_REL];
    __shared__ int lcur[N_REL];

    const int tid = threadIdx.x;
    if (tid < N_REL) lc[tid] = 0;
    __syncthreads();

    for (int n = tid; n < N_ROWS; n += blockDim.x)
        atomicAdd(&lc[rel[n]], 1);
    __syncthreads();

    if (tid == 0) {
        int acc = 0, t = 0;
        for (int r = 0; r < N_REL; ++r) {
            lofs[r] = acc;
            const int c = lc[r];
            for (int s = 0; s < c; s += 16) {
                tiles[t] = make_int4(r, acc + s, (c - s < 16 ? c - s : 16), 0);
                ++t;
            }
            acc += c;
        }
        *nTiles = t;
    }
    __syncthreads();

    if (tid < N_REL) lcur[tid] = lofs[tid];
    __syncthreads();

    for (int n = tid; n < N_ROWS; n += blockDim.x) {
        const int pos = atomicAdd(&lcur[rel[n]], 1);
        rowIdx[pos] = n;
    }
}

// ---------------------------------------------------------------------------
// Kernel B: one block per 16-row tile (all rows share one relation).
// 8 waves; wave w computes the 16x16 output tile at columns [16w, 16w+16)
// via 32x v_wmma_f32_16x16x4_f32 over K = 128.
// ---------------------------------------------------------------------------
__global__ void __launch_bounds__(256)
rel_gemm_kernel(const float* __restrict__ x,
                const float* __restrict__ w,
                const float* __restrict__ bias,
                float* __restrict__ out,
                const int* __restrict__ nTiles,
                const int4* __restrict__ tiles,
                const int* __restrict__ rowIdx) {
    if ((int)blockIdx.x >= *nTiles) return;      // uniform early exit

    const int4 td   = tiles[blockIdx.x];
    const int r     = td.x;
    const int start = td.y;
    const int rows  = td.z;

    __shared__ __align__(16) int sIdx[16];
    const int tid = threadIdx.x;
    if (tid < 16)
        sIdx[tid] = rowIdx[start + (tid < rows ? tid : 0)];  // pad w/ valid row
    __syncthreads();

    const int lane  = tid & 31;
    const int wave  = tid >> 5;          // 0..7
    const int obase = wave * 16;         // output-column base for this wave
    const int half  = lane >> 4;         // 0: K+{0,1}, 1: K+{2,3}
    const int l15   = lane & 15;         // A: row id; B/C/D: column id

    // Row index for this lane's A fragment (gathered x row).
    const int rowA = sIdx[l15];
    // Row indices for this lane's 8 C/D elements (m = half*8 + j),
    // pulled out of LDS once as two b128 reads -> pure-register epilogue.
    const int4 iA = *(const int4*)&sIdx[half * 8];
    const int4 iB = *(const int4*)&sIdx[half * 8 + 4];
    const int rowM[8] = { iA.x, iA.y, iA.z, iA.w, iB.x, iB.y, iB.z, iB.w };

    const float* __restrict__ wr   = w + (size_t)r * (DIM * DIM);
    const float* __restrict__ xrow = x  + (size_t)rowA * DIM + half * 2;
    const float* __restrict__ wcol = wr + (size_t)(half * 2) * DIM + obase + l15;

    v8f c = {};
#pragma unroll 8
    for (int k = 0; k < DIM; k += 4) {
        // A 16x4: lane<16 -> x[row][k..k+1], lane>=16 -> x[row][k+2..k+3]
        v2f a = *(const v2f*)(xrow + k);           // 8B-aligned float2
        // B 4x16: lane<16 -> w[k..k+1][col], lane>=16 -> w[k+2..k+3][col]
        v2f b;
        b.x = wcol[(size_t)k * DIM];
        b.y = wcol[(size_t)k * DIM + DIM];
        c = __builtin_amdgcn_wmma_f32_16x16x4_f32(
                /*neg_a=*/false, a, /*neg_b=*/false, b,
                /*c_mod=*/(short)0, c, /*reuse_a=*/false, /*reuse_b=*/false);
    }

    const float bv     = bias[r * DIM + obase + l15];
    const int   colOff = obase + l15;

    // C/D layout: VGPR j holds row (half*8 + j), col l15.
    if (rows == 16) {
        // Fast path (~94% of tiles): uniform branch, no EXEC masking.
#pragma unroll
        for (int j = 0; j < 8; ++j)
            out[(size_t)rowM[j] * DIM + colOff] = c[j] + bv;
    } else {
#pragma unroll
        for (int j = 0; j < 8; ++j) {
            if (half * 8 + j < rows)
                out[(size_t)rowM[j] * DIM + colOff] = c[j] + bv;
        }
    }
}

extern "C" void kernel_launch(void* const* d_in, const int* in_sizes, int n_in,
                              void* d_out, int out_size, void* d_ws, size_t ws_size,
                              hipStream_t stream) {
    const float* x    = (const float*)d_in[0];
    const int*   rel  = (const int*)d_in[1];
    const float* w    = (const float*)d_in[2];
    const float* bias = (const float*)d_in[3];
    float*       out  = (float*)d_out;

    // Workspace layout (all rebuilt every call; no cross-call state):
    //   [0,16)                      : nTiles (int, padded)
    //   [16, 16+1152*16)            : tile descriptors (int4)
    //   [16+18432, +N_ROWS*4)       : bucketed row indices
    char* ws      = (char*)d_ws;
    int*  nTiles  = (int*)ws;
    int4* tiles   = (int4*)(ws + 16);
    int*  rowIdx  = (int*)(ws + 16 + 1152 * sizeof(int4));

    rel_bucket_kernel<<<1, 1024, 0, stream>>>(rel, nTiles, tiles, rowIdx);
    rel_gemm_kernel<<<MAX_TILES, 256, 0, stream>>>(x, w, bias, out,
                                                   nTiles, tiles, rowIdx);
}